// TopDownTreeLSTM_17042430230603
// MI455X (gfx1250) — compile-verified
//
#include <hip/hip_runtime.h>
#include <hip/hip_bf16.h>

typedef float v2f __attribute__((ext_vector_type(2)));
typedef float v4f __attribute__((ext_vector_type(4)));
typedef float v8f __attribute__((ext_vector_type(8)));

#define LVL 16
#define BTREES 8
#define NNODES 2048
#define MDIM 256
#define ROWS_TOTAL (BTREES * NNODES)   // 16384
#define LVL_ROWS 128                    // rows per block
#define GATE_STRIDE (MDIM * MDIM)       // 65536 floats = 256 KB (fits 24-bit ioffset)

__device__ __forceinline__ float sigf(float x) {
    return 1.0f / (1.0f + __expf(-x));
}
__device__ __forceinline__ float tanhfast(float x) {
    // tanh(x) = 1 - 2/(e^{2x}+1); exact limits at +/-inf
    return 1.0f - 2.0f / (__expf(2.0f * x) + 1.0f);
}

// ---------------------------------------------------------------------------
// Root projection: top_c = root_c @ Wrootx^T + brootx ; top_h = root_h @ Wrooth^T + brooth
// grid: 16 blocks (8 for c, 8 for h), 256 threads
// ---------------------------------------------------------------------------
__global__ void treelstm_root_kernel(const float* __restrict__ root_c,
                                     const float* __restrict__ root_h,
                                     const float* __restrict__ Wrx,
                                     const float* __restrict__ brx,
                                     const float* __restrict__ Wrh,
                                     const float* __restrict__ brh,
                                     float* __restrict__ topc,
                                     float* __restrict__ toph) {
    const int b   = blockIdx.x & 7;
    const bool isH = blockIdx.x >= 8;
    const int m   = threadIdx.x;  // 0..255
    const float* v = (isH ? root_h : root_c) + b * MDIM;
    const float* W = isH ? Wrh : Wrx;
    float s = isH ? brh[m] : brx[m];
    #pragma unroll 4
    for (int k = 0; k < MDIM; ++k)
        s += v[k] * W[m * MDIM + k];
    (isH ? toph : topc)[b * MDIM + m] = s;
}

// ---------------------------------------------------------------------------
// One LSTM level. Block = 256 threads (8 waves). Block tile:
//   rows:  LVL_ROWS (=128)  -> wave w owns 16-row tile w
//   cols:  one 16-wide m-chunk, all 4 gates (i,f,o,u) -> 4+4 fp32 WMMA accums
// ifou = X @ Wx^T + Hpar @ Wh^T + bx + bh,  all via V_WMMA_F32_16X16X4_F32
// ---------------------------------------------------------------------------
__global__ __launch_bounds__(256) void treelstm_level_kernel(
    const float* __restrict__ input_h,    // [L,B,N,M]
    const int*   __restrict__ parent_idx, // [L,B,N]
    const float* __restrict__ Wx,         // [1024,256]
    const float* __restrict__ bxv,        // [1024]
    const float* __restrict__ Wh,         // [1024,256]
    const float* __restrict__ bhv,        // [1024]
    const float* __restrict__ topc,       // [B,M]
    const float* __restrict__ toph,       // [B,M]
    float*       __restrict__ out,        // [L,B,N,2M]
    int level) {
    extern __shared__ float smem[];
    float* Xs = smem;                       // LVL_ROWS*256
    float* Hs = smem + LVL_ROWS * MDIM;     // LVL_ROWS*256
    unsigned* offc = (unsigned*)(smem + 2 * LVL_ROWS * MDIM);  // [LVL_ROWS]
    unsigned* offh = offc + LVL_ROWS;                          // [LVL_ROWS]

    const int tid     = threadIdx.x;
    const int mbase   = blockIdx.x * 16;        // m-chunk
    const int rowbase = blockIdx.y * LVL_ROWS;  // global row group

    // ---- Phase 1: per-row parent source offsets (element offsets) ----
    if (tid < LVL_ROWS) {
        const int Rg = rowbase + tid;       // global row = b*N + n
        const int b  = Rg >> 11;            // N = 2048
        if (level == 0) {
            offc[tid] = (unsigned)(b * MDIM);   // into topc
            offh[tid] = (unsigned)(b * MDIM);   // into toph
        } else {
            const int p = parent_idx[level * ROWS_TOTAL + Rg];
            const unsigned base =
                (unsigned)((level - 1) * ROWS_TOTAL + (b << 11) + p) * 512u;
            offc[tid] = base;          // c section of prev level
            offh[tid] = base + 256u;   // h section of prev level
        }
    }
    __syncthreads();

    // ---- Phase 2: stage X rows (coalesced) and gathered parent-h rows ----
    {
        const v4f* src4 = (const v4f*)(input_h +
                          ((size_t)level * ROWS_TOTAL + rowbase) * MDIM);
        v4f* Xs4 = (v4f*)Xs;
        #pragma unroll 4
        for (int i = tid; i < LVL_ROWS * 64; i += 256)
            Xs4[i] = src4[i];

        const float* hsrc = (level == 0) ? toph : out;
        v4f* Hs4 = (v4f*)Hs;
        #pragma unroll 2
        for (int i = tid; i < LVL_ROWS * 64; i += 256) {
            const int row = i >> 6;
            const int q   = i & 63;
            Hs4[i] = *(const v4f*)(hsrc + offh[row] + q * 4);
        }
    }
    __syncthreads();

    // ---- Phase 3: WMMA K-loop ----
    const int lane = tid & 31;
    const int wave = tid >> 5;    // row tile id, 0..7
    const int col  = lane & 15;   // N index within 16x16 tile
    const int hl   = lane >> 4;   // lane half -> K sub-offset

    v8f aX[4], aH[4];             // per-gate accumulators (x path holds bias)
    #pragma unroll
    for (int g = 0; g < 4; ++g) {
        const int j = g * MDIM + mbase + col;       // output column (gate g)
        const float bsum = bxv[j] + bhv[j];
        #pragma unroll
        for (int e = 0; e < 8; ++e) { aX[g][e] = bsum; aH[g][e] = 0.0f; }
    }

    // A fragment: lane (col) = row M, VGPR pair = K {2hl, 2hl+1}
    const int afrag = (wave * 16 + col) * MDIM + hl * 2;
    const int wfrag = (mbase + col) * MDIM + hl * 2;   // gate 0 column base
    const float* xptr = Xs + afrag;
    const float* hptr = Hs + afrag;
    const float* wxp  = Wx + wfrag;   // gate g at +g*GATE_STRIDE (imm offset)
    const float* whp  = Wh + wfrag;

    #pragma unroll 4
    for (int kk = 0; kk < MDIM / 4; ++kk) {
        const v2f ax = *(const v2f*)(xptr + kk * 4);
        const v2f ah = *(const v2f*)(hptr + kk * 4);
        #pragma unroll
        for (int g = 0; g < 4; ++g) {
            const v2f wb = *(const v2f*)(wxp + kk * 4 + g * GATE_STRIDE);
            aX[g] = __builtin_amdgcn_wmma_f32_16x16x4_f32(
                false, ax, false, wb, (short)0, aX[g], false, false);
        }
        #pragma unroll
        for (int g = 0; g < 4; ++g) {
            const v2f wb = *(const v2f*)(whp + kk * 4 + g * GATE_STRIDE);
            aH[g] = __builtin_amdgcn_wmma_f32_16x16x4_f32(
                false, ah, false, wb, (short)0, aH[g], false, false);
        }
    }

    // ---- Phase 4: gates + cell update + store ----
    // C/D layout: VGPR e holds rows {e (lanes0-15), e+8 (lanes16-31)}, col = lane&15
    const float* csrc = (level == 0) ? topc : out;
    const size_t obase =
        ((size_t)level * ROWS_TOTAL + (size_t)rowbase) * 512 + mbase + col;
    #pragma unroll
    for (int e = 0; e < 8; ++e) {
        const int rloc = wave * 16 + e + hl * 8;       // row within block
        const float iv = sigf(aX[0][e] + aH[0][e]);
        const float fv = sigf(aX[1][e] + aH[1][e]);
        const float ov = sigf(aX[2][e] + aH[2][e]);
        const float uv = sigf(aX[3][e] + aH[3][e]);
        const float pc = csrc[offc[rloc] + mbase + col];
        const float c  = iv * uv + fv * pc;
        const float h  = ov * tanhfast(c);
        out[obase + (size_t)rloc * 512]       = c;
        out[obase + (size_t)rloc * 512 + 256] = h;
    }
}

// ---------------------------------------------------------------------------
extern "C" void kernel_launch(void* const* d_in, const int* in_sizes, int n_in,
                              void* d_out, int out_size, void* d_ws, size_t ws_size,
                              hipStream_t stream) {
    (void)in_sizes; (void)n_in; (void)out_size; (void)ws_size;
    const float* input_h    = (const float*)d_in[0];
    const float* root_c     = (const float*)d_in[1];
    const float* root_h     = (const float*)d_in[2];
    const int*   parent_idx = (const int*)  d_in[3];
    const float* Wrx        = (const float*)d_in[4];
    const float* brx        = (const float*)d_in[5];
    const float* Wrh        = (const float*)d_in[6];
    const float* brh        = (const float*)d_in[7];
    const float* Wx         = (const float*)d_in[8];
    const float* bx         = (const float*)d_in[9];
    const float* Wh         = (const float*)d_in[10];
    const float* bh         = (const float*)d_in[11];
    float* out  = (float*)d_out;
    float* topc = (float*)d_ws;                 // [B,M]
    float* toph = topc + BTREES * MDIM;         // [B,M]

    treelstm_root_kernel<<<16, 256, 0, stream>>>(root_c, root_h, Wrx, brx,
                                                 Wrh, brh, topc, toph);

    const size_t shmem = (2 * LVL_ROWS * MDIM + 2 * LVL_ROWS) * sizeof(float);
    dim3 grid(MDIM / 16, ROWS_TOTAL / LVL_ROWS);  // 16 x 128
    for (int l = 0; l < LVL; ++l) {
        treelstm_level_kernel<<<grid, 256, shmem, stream>>>(
            input_h, parent_idx, Wx, bx, Wh, bh, topc, toph, out, l);
    }
}